// LocalNormalizedCrossCorrelation_33509334843676
// MI455X (gfx1250) — compile-verified
//
#include <hip/hip_runtime.h>

// LNCC, window 9, reflect padding, inputs (2,1,128,128,128) f32, scalar output.
// Fused single pass over z-planes:
//  - contiguous clamped 12x40 window per tile streamed via CDNA5 async
//    global->LDS B128 copies (ASYNCcnt), double buffered (2 issues/thread/plane)
//  - reflect padding resolved at LDS-read time with per-thread precomputed,
//    z-invariant row/column offsets (zero hot-loop cost)
//  - separable 9-tap box sums: x in LDS stage, y per-thread, z via 9-deep LDS
//    ring + running sum (2 flops instead of 9)
//  - per-block partials in d_ws, deterministic 1-block final reduction

#define WIN      9
#define PADR     4
#define DD       128
#define HH       128
#define WW       128
#define NBATCH   2
#define TY       4
#define TX       32
#define TZ       32
#define RY       (TY + 2 * PADR)          // 12
#define RX       (TX + 2 * PADR)          // 40
#define NTHREADS 128
#define PLANE_ELEMS (RY * RX)             // 480 per input
#define NVEC     (2 * RY * (RX / 4))      // 240 b128 loads per plane (f+w)
#define VSLOTS   2                        // ceil(240/128), padded to uniform
#define NBLOCKS  (NBATCH * (DD / TZ) * (HH / TY) * (WW / TX))  // 1024
#define EPSV     1e-8f

__device__ __forceinline__ int reflect_idx(int i) {
    i = (i < 0) ? -i : i;            // numpy 'reflect' (no edge repeat), pad=4<128
    return (i > 127) ? (254 - i) : i;
}

// CDNA5 async global->LDS 16B copy: per-lane 64-bit global address, per-lane
// LDS byte address in VDST. Tracked by ASYNCcnt, in-order completion per wave.
__device__ __forceinline__ void async_copy_b128(unsigned int lds_byte_addr,
                                                const float* gptr) {
    asm volatile("global_load_async_to_lds_b128 %0, %1, off"
                 :: "v"(lds_byte_addr),
                    "v"((unsigned long long)(uintptr_t)gptr)
                 : "memory");
}

__device__ __forceinline__ void wait_async_le2() {
    asm volatile("s_wait_asynccnt 0x2" ::: "memory");
}
__device__ __forceinline__ void wait_async_le0() {
    asm volatile("s_wait_asynccnt 0x0" ::: "memory");
}

__global__ __launch_bounds__(NTHREADS)
void lncc_partial_kernel(const float* __restrict__ fixed,
                         const float* __restrict__ warped,
                         float* __restrict__ partial)
{
    __shared__ alignas(16) float plane[2][2 * PLANE_ELEMS]; // double-buffered
    __shared__ float sx[5][RY][TX];                // x-filtered stage
    __shared__ float ring[5][WIN][TY][TX];         // last 9 xy-filtered planes
    __shared__ float red[NTHREADS];

    const int t  = threadIdx.x;
    int bid = blockIdx.x;
    const int xt = bid & 3;  bid >>= 2;            // 4 x-tiles
    const int yt = bid & 31; bid >>= 5;            // 32 y-tiles
    const int zc = bid & 3;  bid >>= 2;            // 4 z-chunks
    const int b  = bid;                            // 2 batches

    const int x0 = xt * TX;
    const int y0 = yt * TY;
    const int zb = zc * TZ;
    const int yl = t >> 5;                         // 0..3
    const int xl = t & 31;                         // 0..31

    // Clamped contiguous load window; reflected coords always land inside it.
    int ys = y0 - PADR; ys = (ys < 0) ? 0 : ys; ys = (ys > HH - RY) ? HH - RY : ys;
    int xs = x0 - PADR; xs = (xs < 0) ? 0 : xs; xs = (xs > WW - RX) ? WW - RX : xs;

    const float* fixedB  = fixed  + (size_t)b * DD * HH * WW;
    const float* warpedB = warped + (size_t)b * DD * HH * WW;

    for (int j = t; j < 5 * WIN * TY * TX; j += NTHREADS)
        (&ring[0][0][0][0])[j] = 0.0f;

    // B128 slot tables: 240 vector loads/plane, 2 per thread (padded uniform).
    // (y,x) window is z-invariant: per plane we only add zg*H*W.
    const float* slot_base[VSLOTS];
    unsigned int slot_lds[VSLOTS];
#pragma unroll
    for (int s = 0; s < VSLOTS; ++s) {
        int ld  = s * NTHREADS + t;
        int l   = (ld < NVEC) ? ld : 0;            // pad slots: benign dup write
        int q2  = l / (RY * (RX / 4));             // 0=fixed, 1=warped
        int rem = l - q2 * (RY * (RX / 4));
        int r   = rem / (RX / 4);
        int seg = rem - r * (RX / 4);
        slot_base[s] = (q2 ? warpedB : fixedB)
                     + (size_t)(ys + r) * WW + (xs + seg * 4);
        slot_lds[s]  = (unsigned int)(q2 * PLANE_ELEMS + r * RX + seg * 4) * 4u;
    }

    // Per-thread z-invariant reflect remap for the x-filter stage.
    // x-filter slot k handles padded row r = 4k + yl, columns xl..xl+8.
    int colOff[WIN];
#pragma unroll
    for (int d = 0; d < WIN; ++d)
        colOff[d] = reflect_idx(x0 - PADR + xl + d) - xs;
    int rowBase[3];
#pragma unroll
    for (int k = 0; k < 3; ++k)
        rowBase[k] = (reflect_idx(y0 - PADR + 4 * k + yl) - ys) * RX;

    // Prologue: async-load padded plane 0 into buffer 0.
    {
        int zg = reflect_idx(zb - PADR);
        unsigned int ldsBase = (unsigned int)(uintptr_t)(&plane[0][0]);
        size_t zoff = (size_t)zg * HH * WW;
#pragma unroll
        for (int s = 0; s < VSLOTS; ++s)
            async_copy_b128(ldsBase + slot_lds[s], slot_base[s] + zoff);
    }

    float zsum[5] = {0.f, 0.f, 0.f, 0.f, 0.f};
    float acc = 0.0f;
    const int NPLANES = TZ + 2 * PADR;             // 40

    for (int i = 0; i < NPLANES; ++i) {
        // Prefetch plane i+1 into the other buffer, then wait for plane i.
        if (i + 1 < NPLANES) {
            int zg = reflect_idx(zb - PADR + i + 1);
            unsigned int ldsBase =
                (unsigned int)(uintptr_t)(&plane[(i + 1) & 1][0]);
            size_t zoff = (size_t)zg * HH * WW;
#pragma unroll
            for (int s = 0; s < VSLOTS; ++s)
                async_copy_b128(ldsBase + slot_lds[s], slot_base[s] + zoff);
            wait_async_le2();                      // in-order: plane i done
        } else {
            wait_async_le0();
        }
        __syncthreads();

        // x-filter: 12 rows x 32 cols x 5 quantities (3 row-slots/thread).
        const float* pf = &plane[i & 1][0];
        const float* pw = pf + PLANE_ELEMS;
#pragma unroll
        for (int k = 0; k < 3; ++k) {
            const int r = 4 * k + yl;              // padded row 0..11
            const float* fr = pf + rowBase[k];
            const float* wr = pw + rowBase[k];
            float sf = 0.f, sw = 0.f, sff = 0.f, sww = 0.f, sfw = 0.f;
#pragma unroll
            for (int d = 0; d < WIN; ++d) {
                float fv = fr[colOff[d]];
                float wv = wr[colOff[d]];
                sf += fv;
                sw += wv;
                sff = fmaf(fv, fv, sff);
                sww = fmaf(wv, wv, sww);
                sfw = fmaf(fv, wv, sfw);
            }
            sx[0][r][xl] = sf;  sx[1][r][xl] = sw;  sx[2][r][xl] = sff;
            sx[3][r][xl] = sww; sx[4][r][xl] = sfw;
        }
        __syncthreads();

        // y-filter + z running sum via 9-deep LDS ring (thread-private slot).
        int head = i % WIN;
#pragma unroll
        for (int q = 0; q < 5; ++q) {
            float s = 0.f;
#pragma unroll
            for (int j = 0; j < WIN; ++j) s += sx[q][yl + j][xl];
            float old = ring[q][head][yl][xl];
            ring[q][head][yl][xl] = s;
            zsum[q] += s - old;
        }

        if (i >= WIN - 1) {                        // emit z = zb + i - 8
            const float inv = 1.0f / 729.0f;
            float mu_f   = zsum[0] * inv;
            float mu_w   = zsum[1] * inv;
            float sig_f  = fmaxf(zsum[2] * inv - mu_f * mu_f, EPSV);
            float sig_w  = fmaxf(zsum[3] * inv - mu_w * mu_w, EPSV);
            float sig_fw = zsum[4] * inv - mu_f * mu_w;
            float denom  = fmaxf(sqrtf(fmaxf(sig_f * sig_w, EPSV * EPSV)), EPSV);
            float v = sig_fw / denom;
            v = fminf(fmaxf(v, -10.0f), 10.0f);
            acc += v;
        }
    }

    // Block reduction -> one partial per block (deterministic).
    red[t] = acc;
    __syncthreads();
#pragma unroll
    for (int sft = NTHREADS >> 1; sft > 0; sft >>= 1) {
        if (t < sft) red[t] += red[t + sft];
        __syncthreads();
    }
    if (t == 0) partial[blockIdx.x] = red[0];
}

__global__ __launch_bounds__(NTHREADS)
void lncc_finalize_kernel(const float* __restrict__ partial,
                          float* __restrict__ out)
{
    __shared__ float red[NTHREADS];
    float s = 0.0f;
    for (int j = threadIdx.x; j < NBLOCKS; j += NTHREADS) s += partial[j];
    red[threadIdx.x] = s;
    __syncthreads();
    for (int sft = NTHREADS >> 1; sft > 0; sft >>= 1) {
        if (threadIdx.x < sft) red[threadIdx.x] += red[threadIdx.x + sft];
        __syncthreads();
    }
    if (threadIdx.x == 0)
        out[0] = -red[0] / (float)((size_t)NBATCH * DD * HH * WW);
}

extern "C" void kernel_launch(void* const* d_in, const int* in_sizes, int n_in,
                              void* d_out, int out_size, void* d_ws, size_t ws_size,
                              hipStream_t stream) {
    (void)in_sizes; (void)n_in; (void)out_size; (void)ws_size;
    const float* fixed  = (const float*)d_in[0];
    const float* warped = (const float*)d_in[1];
    float* out     = (float*)d_out;
    float* partial = (float*)d_ws;   // needs NBLOCKS floats (4 KB)

    lncc_partial_kernel<<<NBLOCKS, NTHREADS, 0, stream>>>(fixed, warped, partial);
    lncc_finalize_kernel<<<1, NTHREADS, 0, stream>>>(partial, out);
}